// MyGATlayer_2568390443567
// MI455X (gfx1250) — compile-verified
//
#include <hip/hip_runtime.h>
#include <hip/hip_bf16.h>
#include <math.h>

typedef float v2f __attribute__((ext_vector_type(2)));
typedef float v8f __attribute__((ext_vector_type(8)));

#define IN_FEATS 128
#define HF 64   // HEADS * OUT_FEATS
#define HEADS 4
#define OUTF 16

// ---------------------------------------------------------------------------
// Kernel 1: h[N,64] = x[N,128] @ W[64,128]^T  using V_WMMA_F32_16X16X4_F32.
// Block = 128 threads = 4 waves; wave w computes the 16x16 tile for head w.
// ---------------------------------------------------------------------------
__global__ void gat_gemm_wmma(const float* __restrict__ x,
                              const float* __restrict__ W,
                              float* __restrict__ h, int N) {
    const int wave = threadIdx.x >> 5;
    const int lane = threadIdx.x & 31;
    const int m0   = blockIdx.x * 16;
    const int n0   = wave * 16;                 // output-feature tile == head
    int row = m0 + (lane & 15);
    if (row >= N) row = N - 1;                  // clamp (keeps EXEC all-ones)
    const int col   = n0 + (lane & 15);
    const int khalf = (lane >> 4) * 2;          // lanes 0-15: K=0,1  lanes 16-31: K=2,3

    const float* __restrict__ xr = x + (size_t)row * IN_FEATS;
    const float* __restrict__ wr = W + (size_t)col * IN_FEATS;

    v8f acc = {};
    #pragma unroll
    for (int k = 0; k < IN_FEATS; k += 4) {
        v2f a, b;
        a.x = xr[k + khalf];
        a.y = xr[k + khalf + 1];
        b.x = wr[k + khalf];
        b.y = wr[k + khalf + 1];
        acc = __builtin_amdgcn_wmma_f32_16x16x4_f32(
            /*neg_a=*/false, a, /*neg_b=*/false, b,
            /*c_mod=*/(short)0, acc, /*reuse_a=*/false, /*reuse_b=*/false);
    }

    // C/D layout: VGPR r -> M = r (lanes 0-15) or M = 8+r (lanes 16-31); N = lane&15
    const int rbase = m0 + ((lane >> 4) * 8);
    #pragma unroll
    for (int r = 0; r < 8; ++r) {
        const int m = rbase + r;
        if (m < N) h[(size_t)m * HF + col] = acc[r];
    }
}

// ---------------------------------------------------------------------------
// Kernel 2: per-(node,head) attention scalars ai = <h, att_i>, aj = <h, att_j>
// ---------------------------------------------------------------------------
__global__ void gat_node_att(const float* __restrict__ h,
                             const float* __restrict__ att_i,
                             const float* __restrict__ att_j,
                             float* __restrict__ ai, float* __restrict__ aj,
                             int NH) {
    const int t = blockIdx.x * blockDim.x + threadIdx.x;
    if (t >= NH) return;
    const int hd = t & (HEADS - 1);
    const float* hp = h + (size_t)t * OUTF;     // t = n*HEADS + hd -> h + n*64 + hd*16
    float si = 0.f, sj = 0.f;
    #pragma unroll
    for (int f = 0; f < OUTF; ++f) {
        const float v = hp[f];
        si += v * att_i[hd * OUTF + f];
        sj += v * att_j[hd * OUTF + f];
    }
    ai[t] = si;
    aj[t] = sj;
}

// Monotone float<->u32 encoding for atomic max (handles all sign combos).
__device__ __forceinline__ unsigned enc_f32(float f) {
    unsigned u = __float_as_uint(f);
    return (u & 0x80000000u) ? ~u : (u | 0x80000000u);
}
__device__ __forceinline__ float dec_f32(unsigned u) {
    return __uint_as_float((u & 0x80000000u) ? (u ^ 0x80000000u) : ~u);
}

// ---------------------------------------------------------------------------
// Kernel 3: per-(edge,head) logits + leaky_relu + segment atomic-max over dst
// ---------------------------------------------------------------------------
__global__ void gat_edge_logits(const int* __restrict__ src,
                                const int* __restrict__ dst,
                                const float* __restrict__ ai,
                                const float* __restrict__ aj,
                                float* __restrict__ alpha,
                                unsigned* __restrict__ amax_enc, int EH) {
    const int t = blockIdx.x * blockDim.x + threadIdx.x;
    if (t >= EH) return;
    const int e  = t >> 2;
    const int hd = t & (HEADS - 1);
    const int s = src[e];
    const int d = dst[e];
    float a = ai[d * HEADS + hd] + aj[s * HEADS + hd];
    a = (a > 0.f) ? a : 0.01f * a;              // leaky_relu, slope 0.01
    alpha[t] = a;
    atomicMax(&amax_enc[d * HEADS + hd], enc_f32(a));
}

// ---------------------------------------------------------------------------
// Kernel 4: decode amax in place; enc==0 (no edges) decodes to NaN -> 0.0
// (matches reference's where(isfinite(amax), amax, 0))
// ---------------------------------------------------------------------------
__global__ void gat_fix_amax(unsigned* __restrict__ amax_enc, int NH) {
    const int t = blockIdx.x * blockDim.x + threadIdx.x;
    if (t >= NH) return;
    const float v = dec_f32(amax_enc[t]);
    float* fp = reinterpret_cast<float*>(amax_enc);
    fp[t] = isfinite(v) ? v : 0.f;
}

// ---------------------------------------------------------------------------
// Kernel 5: ex = exp(alpha - amax[dst]); denom[dst] += ex (atomic)
// ---------------------------------------------------------------------------
__global__ void gat_edge_exp(const int* __restrict__ dst,
                             const float* __restrict__ amax,
                             float* __restrict__ alpha,
                             float* __restrict__ denom, int EH) {
    const int t = blockIdx.x * blockDim.x + threadIdx.x;
    if (t >= EH) return;
    const int e  = t >> 2;
    const int hd = t & (HEADS - 1);
    const int d  = dst[e];
    const float ex = expf(alpha[t] - amax[d * HEADS + hd]);
    alpha[t] = ex;
    atomicAdd(&denom[d * HEADS + hd], ex);
}

// ---------------------------------------------------------------------------
// Kernel 6: weighted scatter-sum.  One thread per (edge, 4-float chunk):
// 16 threads per edge cover 64 features; float4 gather from h (L2-resident),
// 4x global_atomic_add_f32 scatter to out.
// ---------------------------------------------------------------------------
__global__ void gat_scatter(const int* __restrict__ src,
                            const int* __restrict__ dst,
                            const float* __restrict__ h,
                            const float* __restrict__ ex,
                            const float* __restrict__ denom,
                            float* __restrict__ out, int E16) {
    const int t = blockIdx.x * blockDim.x + threadIdx.x;
    if (t >= E16) return;
    const int e  = t >> 4;
    const int q  = t & 15;       // 4-float chunk index within the 64 features
    const int hd = q >> 2;       // head = (q*4)/16
    const int s = src[e];
    const int d = dst[e];
    const float w = ex[e * HEADS + hd] / (denom[d * HEADS + hd] + 1e-16f);
    const float4 v = *reinterpret_cast<const float4*>(h + (size_t)s * HF + q * 4);
    float* op = out + (size_t)d * HF + q * 4;
    atomicAdd(op + 0, v.x * w);
    atomicAdd(op + 1, v.y * w);
    atomicAdd(op + 2, v.z * w);
    atomicAdd(op + 3, v.w * w);
}

extern "C" void kernel_launch(void* const* d_in, const int* in_sizes, int n_in,
                              void* d_out, int out_size, void* d_ws, size_t ws_size,
                              hipStream_t stream) {
    (void)n_in; (void)out_size; (void)ws_size;
    const float* x     = (const float*)d_in[0];
    const int*   ei    = (const int*)d_in[1];
    const float* W     = (const float*)d_in[2];
    const float* att_i = (const float*)d_in[3];
    const float* att_j = (const float*)d_in[4];
    float* out = (float*)d_out;

    const int N = in_sizes[0] / IN_FEATS;
    const int E = in_sizes[1] / 2;
    const int* src = ei;          // edge_index[0] = j (sender)
    const int* dst = ei + E;      // edge_index[1] = i (softmax segment)

    // workspace layout (fp32 elements)
    float*    h     = (float*)d_ws;                        // N*64
    float*    ai    = h  + (size_t)N * HF;                 // N*4
    float*    aj    = ai + (size_t)N * HEADS;              // N*4
    unsigned* amax  = (unsigned*)(aj + (size_t)N * HEADS); // N*4 (enc u32 -> f32)
    float*    denom = (float*)(amax + (size_t)N * HEADS);  // N*4
    float*    alpha = denom + (size_t)N * HEADS;           // E*4

    const int NH  = N * HEADS;
    const int EH  = E * HEADS;
    const int E16 = E * 16;

    // zero-init: out, amax(enc, 0 == -inf sentinel) and denom are contiguous
    hipMemsetAsync(out, 0, (size_t)N * HF * sizeof(float), stream);
    hipMemsetAsync(amax, 0, (size_t)NH * sizeof(unsigned) * 2, stream);

    gat_gemm_wmma<<<(N + 15) / 16, 128, 0, stream>>>(x, W, h, N);
    gat_node_att<<<(NH + 255) / 256, 256, 0, stream>>>(h, att_i, att_j, ai, aj, NH);
    gat_edge_logits<<<(EH + 255) / 256, 256, 0, stream>>>(src, dst, ai, aj, alpha, amax, EH);
    gat_fix_amax<<<(NH + 255) / 256, 256, 0, stream>>>(amax, NH);
    gat_edge_exp<<<(EH + 255) / 256, 256, 0, stream>>>(dst, (const float*)amax, alpha, denom, EH);
    gat_scatter<<<(E16 + 255) / 256, 256, 0, stream>>>(src, dst, h, alpha, denom, out, E16);
}